// KNet_architecture_v2_21036749816496
// MI455X (gfx1250) — compile-verified
//
#include <hip/hip_runtime.h>
#include <math.h>

// ---------------------------------------------------------------------------
// KalmanNet step on MI455X (gfx1250).
// Memory-bound GEMV chain (~352 MB fp32 weights read once => ~15us floor at
// 23.3 TB/s). All large matvecs run through V_WMMA_F32_16X16X4_F32 used as a
// streaming GEMV: W tile -> A operand, x chunk -> column 0 of B, accumulate C.
// Inner loop is split into a 16-load clause phase then a WMMA phase so each
// wave keeps ~4KB of demand loads in flight; global_prefetch_b8 runs ~1KB
// ahead per lane. 4 independent accumulators hide WMMA latency.
// GRU's two GEMVs (Wih@x, Whh@h) are fused into one launch (gridDim.y == 2).
// FC2b (576 x 46080) is split-K (40 chunks) + deterministic reduction.
// ---------------------------------------------------------------------------

typedef __attribute__((ext_vector_type(2))) float v2f;
typedef __attribute__((ext_vector_type(8))) float v8f;

#define WAVES_PER_BLOCK 8
#define TB (WAVES_PER_BLOCK * 32)

// Core streaming loop: one wave, 16 rows of W, K multiple of 32.
// A 16x4 f32 layout: lanes 0-15 = rows, VGPR0/1 = K0/K1; lanes 16-31 = K2/K3.
// B  4x16 f32 layout: column n in lanes n (k=0,1) and n+16 (k=2,3); we only
// populate column 0 (lanes 0 and 16), other lanes supply zeros via cndmask.
__device__ __forceinline__ v8f wmma_gemv_body(const float* __restrict__ wrow,
                                              const float* __restrict__ xp,
                                              int Klen, bool col0) {
  v8f acc0 = {0.f, 0.f, 0.f, 0.f, 0.f, 0.f, 0.f, 0.f};
  v8f acc1 = acc0, acc2 = acc0, acc3 = acc0;
  for (int k = 0; k < Klen; k += 32) {
    __builtin_prefetch(wrow + k + 256, 0, 1);  // global_prefetch_b8, ~1KB ahead
    // Phase 1: issue all 16 b64 loads as one clause (deep MLP per wave).
    v2f a[8], xv[8];
#pragma unroll
    for (int u = 0; u < 8; ++u) {
      a[u]  = *(const v2f*)(wrow + k + 4 * u);  // 2 consecutive W elems
      xv[u] = *(const v2f*)(xp + k + 4 * u);    // matching x chunk
    }
    // Phase 2: drain into 4 independent WMMA accumulation chains.
#pragma unroll
    for (int u = 0; u < 8; ++u) {
      v2f b;
      b.x = col0 ? xv[u].x : 0.0f;             // v_cndmask, EXEC untouched
      b.y = col0 ? xv[u].y : 0.0f;
      if ((u & 3) == 0)
        acc0 = __builtin_amdgcn_wmma_f32_16x16x4_f32(false, a[u], false, b,
                                                     (short)0, acc0, false, false);
      else if ((u & 3) == 1)
        acc1 = __builtin_amdgcn_wmma_f32_16x16x4_f32(false, a[u], false, b,
                                                     (short)0, acc1, false, false);
      else if ((u & 3) == 2)
        acc2 = __builtin_amdgcn_wmma_f32_16x16x4_f32(false, a[u], false, b,
                                                     (short)0, acc2, false, false);
      else
        acc3 = __builtin_amdgcn_wmma_f32_16x16x4_f32(false, a[u], false, b,
                                                     (short)0, acc3, false, false);
    }
  }
  return (acc0 + acc1) + (acc2 + acc3);
}

// Common per-wave tile setup + epilogue.
template <bool RELU>
__device__ __forceinline__ void gemv_tile(const float* __restrict__ W,
                                          const float* __restrict__ x,
                                          const float* __restrict__ bias,
                                          float* __restrict__ y,
                                          int row0, int K) {
  const int lane = threadIdx.x & 31;
  const int r    = lane & 15;
  const int half = lane >> 4;                  // K-half: 0 -> k0/k1, 1 -> k2/k3
  const float* wrow = W + (size_t)(row0 + r) * (size_t)K + 2 * half;
  const float* xp   = x + 2 * half;
  v8f acc = wmma_gemv_body(wrow, xp, K, r == 0);
  // D column 0: lane 0 -> m=row0..row0+7 ; lane 16 -> m=row0+8..row0+15
  if (r == 0) {
    const int mb = row0 + 8 * half;
#pragma unroll
    for (int i = 0; i < 8; ++i) {
      float v = acc[i] + bias[mb + i];
      if (RELU) v = fmaxf(v, 0.0f);
      y[mb + i] = v;
    }
  }
}

// y[m] = act(W[m,:] @ x + bias[m]); one wave per 16-row tile.
// Requires: M % 16 == 0, K % 32 == 0.
template <bool RELU>
__global__ __launch_bounds__(TB) void gemv_wmma(
    const float* __restrict__ W, const float* __restrict__ x,
    const float* __restrict__ bias, float* __restrict__ y, int M, int K) {
  const int tile = blockIdx.x * WAVES_PER_BLOCK + (threadIdx.x >> 5);
  const int row0 = tile << 4;
  if (row0 >= M) return;                       // wave-uniform
  gemv_tile<RELU>(W, x, bias, y, row0, K);
}

// Fused GRU pre-activation: gridDim.y==2 selects gi = Wih@xi + bih (y==0)
// or gh = Whh@h + bhh (y==1). Both matrices are [M x K*], M % 16 == 0.
__global__ __launch_bounds__(TB) void gru_gemv2(
    const float* __restrict__ Wih, const float* __restrict__ xi, int Ki,
    const float* __restrict__ bih, float* __restrict__ gi,
    const float* __restrict__ Whh, const float* __restrict__ h, int Kh,
    const float* __restrict__ bhh, float* __restrict__ gh, int M) {
  const int tile = blockIdx.x * WAVES_PER_BLOCK + (threadIdx.x >> 5);
  const int row0 = tile << 4;
  if (row0 >= M) return;
  if (blockIdx.y == 0)
    gemv_tile<false>(Wih, xi, bih, gi, row0, Ki);
  else
    gemv_tile<false>(Whh, h, bhh, gh, row0, Kh);
}

// Split-K partial GEMV: grid.y = split index, each covers Kc (% 32 == 0).
__global__ __launch_bounds__(TB) void gemv_wmma_splitk(
    const float* __restrict__ W, const float* __restrict__ x,
    float* __restrict__ part, int M, int K, int Kc) {
  const int lane = threadIdx.x & 31;
  const int tile = blockIdx.x * WAVES_PER_BLOCK + (threadIdx.x >> 5);
  const int row0 = tile << 4;
  if (row0 >= M) return;
  const int r    = lane & 15;
  const int half = lane >> 4;
  const int kb   = blockIdx.y * Kc;
  const float* wrow = W + (size_t)(row0 + r) * (size_t)K + kb + 2 * half;
  const float* xp   = x + kb + 2 * half;
  v8f acc = wmma_gemv_body(wrow, xp, Kc, r == 0);
  if (r == 0) {
    const int mb = row0 + 8 * half;
#pragma unroll
    for (int i = 0; i < 8; ++i)
      part[(size_t)blockIdx.y * M + mb + i] = acc[i];
  }
}

// y[m] = bias[m] + sum_s part[s*M + m]   (deterministic split-K reduction)
__global__ void reduce_bias(const float* __restrict__ part,
                            const float* __restrict__ bias,
                            float* __restrict__ y, int M, int S) {
  int m = blockIdx.x * blockDim.x + threadIdx.x;
  if (m >= M) return;
  float s = bias[m];
  for (int i = 0; i < S; ++i) s += part[(size_t)i * M + m];
  y[m] = s;
}

// FC5 (480x24), FC6 (480x24), FC7 (960x48) fused; thread per output row.
__global__ void fc567(const float* __restrict__ W5, const float* __restrict__ b5,
                      const float* __restrict__ fe,
                      const float* __restrict__ W6, const float* __restrict__ b6,
                      const float* __restrict__ fu,
                      const float* __restrict__ W7, const float* __restrict__ b7,
                      const float* __restrict__ od, const float* __restrict__ oid,
                      float* __restrict__ out5, float* __restrict__ out6,
                      float* __restrict__ out7) {
  int id = blockIdx.x * blockDim.x + threadIdx.x;
  if (id < 480) {
    const float* w = W5 + id * 24;
    float s = b5[id];
#pragma unroll
    for (int j = 0; j < 24; ++j) s += w[j] * fe[j];
    out5[id] = fmaxf(s, 0.0f);
  } else if (id < 960) {
    int m = id - 480;
    const float* w = W6 + m * 24;
    float s = b6[m];
#pragma unroll
    for (int j = 0; j < 24; ++j) s += w[j] * fu[j];
    out6[m] = fmaxf(s, 0.0f);
  } else if (id < 1920) {
    int m = id - 960;
    const float* w = W7 + m * 48;
    float s = b7[m];
#pragma unroll
    for (int j = 0; j < 24; ++j) s += w[j] * od[j];
#pragma unroll
    for (int j = 0; j < 24; ++j) s += w[24 + j] * oid[j];
    out7[m] = fmaxf(s, 0.0f);
  }
}

// PyTorch GRU gates: gi/gh are [3H] = (r,z,n); hn = (1-z)*n + z*h.
__global__ void gru_gate(const float* __restrict__ gi,
                         const float* __restrict__ gh,
                         const float* __restrict__ h,
                         float* __restrict__ hn, int H) {
  int i = blockIdx.x * blockDim.x + threadIdx.x;
  if (i >= H) return;
  float r = 1.0f / (1.0f + expf(-(gi[i] + gh[i])));
  float z = 1.0f / (1.0f + expf(-(gi[i + H] + gh[i + H])));
  float n = tanhf(gi[i + 2 * H] + r * gh[i + 2 * H]);
  hn[i] = (1.0f - z) * n + z * h[i];
}

static inline int tiles_grid(int M) {
  return (M / 16 + WAVES_PER_BLOCK - 1) / WAVES_PER_BLOCK;
}

extern "C" void kernel_launch(void* const* d_in, const int* in_sizes, int n_in,
                              void* d_out, int out_size, void* d_ws, size_t ws_size,
                              hipStream_t stream) {
  (void)in_sizes; (void)n_in; (void)out_size; (void)ws_size;
  const float* obs_diff   = (const float*)d_in[0];
  const float* obs_innov  = (const float*)d_in[1];
  const float* fw_evol    = (const float*)d_in[2];
  const float* fw_update  = (const float*)d_in[3];
  const float* h_Q        = (const float*)d_in[4];
  const float* h_Sigma    = (const float*)d_in[5];
  const float* h_S        = (const float*)d_in[6];
  const float* Wih_Q      = (const float*)d_in[7];
  const float* Whh_Q      = (const float*)d_in[8];
  const float* bih_Q      = (const float*)d_in[9];
  const float* bhh_Q      = (const float*)d_in[10];
  const float* Wih_Sig    = (const float*)d_in[11];
  const float* Whh_Sig    = (const float*)d_in[12];
  const float* bih_Sig    = (const float*)d_in[13];
  const float* bhh_Sig    = (const float*)d_in[14];
  const float* Wih_S      = (const float*)d_in[15];
  const float* Whh_S      = (const float*)d_in[16];
  const float* bih_S      = (const float*)d_in[17];
  const float* bhh_S      = (const float*)d_in[18];
  const float* W1         = (const float*)d_in[19];
  const float* b1         = (const float*)d_in[20];
  const float* W2a        = (const float*)d_in[21];
  const float* b2a        = (const float*)d_in[22];
  const float* W2b        = (const float*)d_in[23];
  const float* b2b        = (const float*)d_in[24];
  // d_in[25..28] = W3/b3/W4/b4: unused by the reference forward pass.
  const float* W5         = (const float*)d_in[29];
  const float* b5         = (const float*)d_in[30];
  const float* W6         = (const float*)d_in[31];
  const float* b6         = (const float*)d_in[32];
  const float* W7         = (const float*)d_in[33];
  const float* b7         = (const float*)d_in[34];
  float* out = (float*)d_out;

  // Workspace layout (floats, all offsets 64B aligned). Total 76800 f = 300KB.
  float* ws   = (float*)d_ws;
  float* xSig = ws;            // [1056] = [hQ (576) | out6 (480)]
  float* out5 = ws + 1056;     // [480]
  float* xS   = ws + 1536;     // [1536] = [out1 (576) | out7 (960)]
  float* in2  = ws + 3072;     // [1152] = [hSig (576) | hS (576)]
  float* gi   = ws + 4224;     // [1728] GRU input gates (reused)
  float* gh   = ws + 5952;     // [1728] GRU hidden gates (reused)
  float* hid  = ws + 7680;     // [46080] FC2 hidden
  float* part = ws + 53760;    // [40*576] split-K partials

  const dim3 gruGrid(tiles_grid(1728), 2);

  // FC5 / FC6 / FC7 (independent small GEMVs)
  fc567<<<8, 256, 0, stream>>>(W5, b5, fw_evol, W6, b6, fw_update,
                               W7, b7, obs_diff, obs_innov,
                               out5, xSig + 576, xS + 576);

  // GRU_Q: hQ -> xSig[0:576]
  gru_gemv2<<<gruGrid, TB, 0, stream>>>(Wih_Q, out5, 480, bih_Q, gi,
                                        Whh_Q, h_Q, 576, bhh_Q, gh, 1728);
  gru_gate<<<3, 256, 0, stream>>>(gi, gh, h_Q, xSig, 576);

  // GRU_Sigma: hSig -> in2[0:576]
  gru_gemv2<<<gruGrid, TB, 0, stream>>>(Wih_Sig, xSig, 1056, bih_Sig, gi,
                                        Whh_Sig, h_Sigma, 576, bhh_Sig, gh, 1728);
  gru_gate<<<3, 256, 0, stream>>>(gi, gh, h_Sigma, in2, 576);

  // FC1: out1 = relu(W1 @ hSig + b1) -> xS[0:576]
  gemv_wmma<true><<<tiles_grid(576), TB, 0, stream>>>(W1, in2, b1, xS, 576, 576);

  // GRU_S: hS -> in2[576:1152]
  gru_gemv2<<<gruGrid, TB, 0, stream>>>(Wih_S, xS, 1536, bih_S, gi,
                                        Whh_S, h_S, 576, bhh_S, gh, 1728);
  gru_gate<<<3, 256, 0, stream>>>(gi, gh, h_S, in2 + 576, 576);

  // FC2a: hid = relu(W2a @ in2 + b2a)   (2880 tiles -> plenty of waves)
  gemv_wmma<true><<<tiles_grid(46080), TB, 0, stream>>>(W2a, in2, b2a, hid, 46080, 1152);

  // FC2b: split-K (40 x 1152) + deterministic reduction with bias
  dim3 g2(tiles_grid(576), 40);
  gemv_wmma_splitk<<<g2, TB, 0, stream>>>(W2b, hid, part, 576, 46080, 1152);
  reduce_bias<<<3, 256, 0, stream>>>(part, b2b, out, 576, 40);
}